// DMoN_1675037246074
// MI455X (gfx1250) — compile-verified
//
#include <hip/hip_runtime.h>
#include <math.h>

// ---------------------------------------------------------------------------
// DMoN forward for MI455X (gfx1250, wave32, WMMA f32 16x16x4)
// N=100000 nodes, C=128 channels, K=16 clusters, E=3.2M edges
// ---------------------------------------------------------------------------

typedef __attribute__((ext_vector_type(2))) float v2f;
typedef __attribute__((ext_vector_type(8))) float v8f;

#define C_DIM 128
#define K_DIM 16

static __device__ __forceinline__ v8f wmma_f32_4(v2f a, v2f b, v8f c) {
  // D = A(16x4,f32) x B(4x16,f32) + C(16x16,f32)
  return __builtin_amdgcn_wmma_f32_16x16x4_f32(false, a, false, b, (short)0, c,
                                               false, false);
}

// ------------------------------ zero scratch -------------------------------
__global__ void zero_kernel(float* __restrict__ p, long long n) {
  long long i = (long long)blockIdx.x * blockDim.x + threadIdx.x;
  long long stride = (long long)gridDim.x * blockDim.x;
  for (; i < n; i += stride) p[i] = 0.0f;
}

// ------------------- assignments = softmax(X @ W^T + b) --------------------
// One wave per 16-node tile. D[node m][cluster n] accumulated over C in
// K=4 chunks via v_wmma_f32_16x16x4_f32, then softmax across lanes (clusters).
__global__ void assign_kernel(const float* __restrict__ feat,
                              const float* __restrict__ fcw,
                              const float* __restrict__ fcb,
                              float* __restrict__ assign, int nTiles) {
  int wave = (int)((blockIdx.x * blockDim.x + threadIdx.x) >> 5);
  if (wave >= nTiles) return;  // wave-uniform; grid sized exactly anyway
  int lane = threadIdx.x & 31;
  int col  = lane & 15;   // A: node row m ; B: cluster n ; D: column n
  int half = lane >> 4;   // selects K pair
  int ko   = half * 2;

  const float* Arow = feat + (size_t)(wave * 16 + col) * C_DIM;  // features row
  const float* Brow = fcw + (size_t)col * C_DIM;                 // fc_w row n

  v8f acc = {0.f, 0.f, 0.f, 0.f, 0.f, 0.f, 0.f, 0.f};
  for (int kb = 0; kb < C_DIM; kb += 4) {
    v2f a = *(const v2f*)(Arow + kb + ko);
    v2f b = *(const v2f*)(Brow + kb + ko);
    acc = wmma_f32_4(a, b, acc);
  }

  float bias = fcb[col];
#pragma unroll
  for (int v = 0; v < 8; ++v) {
    float x = acc[v] + bias;
    // softmax over 16 clusters == across lanes within each 16-lane half
    float m = x;
    m = fmaxf(m, __shfl_xor(m, 1, 32));
    m = fmaxf(m, __shfl_xor(m, 2, 32));
    m = fmaxf(m, __shfl_xor(m, 4, 32));
    m = fmaxf(m, __shfl_xor(m, 8, 32));
    float e = __expf(x - m);
    float s = e;
    s += __shfl_xor(s, 1, 32);
    s += __shfl_xor(s, 2, 32);
    s += __shfl_xor(s, 4, 32);
    s += __shfl_xor(s, 8, 32);
    acc[v] = e / s;
  }

#pragma unroll
  for (int v = 0; v < 8; ++v) {
    int r = v + half * 8;  // node row within tile
    assign[(size_t)(wave * 16 + r) * K_DIM + col] = acc[v];
  }
}

// ------------------------- cluster_sizes = colsum(S) -----------------------
__global__ void cs_kernel(const float* __restrict__ assign,
                          float* __restrict__ cs, int N) {
  __shared__ float s[K_DIM];
  int t = threadIdx.x;
  if (t < K_DIM) s[t] = 0.0f;
  __syncthreads();
  int c = t & 15;
  int stride = (int)((gridDim.x * blockDim.x) >> 4);
  float p = 0.0f;
  for (int i = (int)((blockIdx.x * blockDim.x + t) >> 4); i < N; i += stride)
    p += assign[(size_t)i * K_DIM + c];
  atomicAdd(&s[c], p);
  __syncthreads();
  if (t < K_DIM) atomicAdd(&cs[t], s[t]);
}

// --------------------- degrees + total number_of_edges ---------------------
__global__ void deg_kernel(const int* __restrict__ ecol,
                           const float* __restrict__ eval,
                           float* __restrict__ deg, float* __restrict__ scal,
                           int E) {
  __shared__ float s[8];
  int gid = blockIdx.x * blockDim.x + threadIdx.x;
  int stride = gridDim.x * blockDim.x;
  float p = 0.0f;
  for (int e = gid; e < E; e += stride) {
    float v = eval[e];
    atomicAdd(&deg[ecol[e]], v);
    p += v;
  }
  p += __shfl_xor(p, 1, 32);
  p += __shfl_xor(p, 2, 32);
  p += __shfl_xor(p, 4, 32);
  p += __shfl_xor(p, 8, 32);
  p += __shfl_xor(p, 16, 32);
  if ((threadIdx.x & 31) == 0) s[threadIdx.x >> 5] = p;
  __syncthreads();
  if (threadIdx.x == 0) {
    float t = 0.0f;
    for (int i = 0; i < 8; ++i) t += s[i];
    atomicAdd(&scal[0], t);
  }
}

// ----------------- a_s[row] += val * assignments[col]  (SpMM) --------------
// 16 lanes per edge: lane k handles cluster k of its edge.
__global__ void spmm_kernel(const int* __restrict__ erow,
                            const int* __restrict__ ecol,
                            const float* __restrict__ eval,
                            const float* __restrict__ assign,
                            float* __restrict__ a_s, int E) {
  int idx = blockIdx.x * blockDim.x + threadIdx.x;
  int e = idx >> 4;
  int k = idx & 15;
  if (e < E) {
    int r = erow[e];
    int c = ecol[e];
    atomicAdd(&a_s[(size_t)r * K_DIM + k],
              eval[e] * assign[(size_t)c * K_DIM + k]);
  }
}

// ------------- trace(S^T A S) and nl = S^T degrees (only traces) -----------
__global__ void trace_kernel(const float* __restrict__ a_s,
                             const float* __restrict__ assign,
                             const float* __restrict__ deg,
                             float* __restrict__ nl, float* __restrict__ scal,
                             int N) {
  __shared__ float snl[K_DIM];
  __shared__ float str[8];
  int t = threadIdx.x;
  if (t < K_DIM) snl[t] = 0.0f;
  __syncthreads();
  int c = t & 15;
  int stride = (int)((gridDim.x * blockDim.x) >> 4);
  float ptr = 0.0f, pnl = 0.0f;
  for (int i = (int)((blockIdx.x * blockDim.x + t) >> 4); i < N; i += stride) {
    float a = assign[(size_t)i * K_DIM + c];
    ptr += a_s[(size_t)i * K_DIM + c] * a;
    pnl += a * deg[i];
  }
  atomicAdd(&snl[c], pnl);
  ptr += __shfl_xor(ptr, 1, 32);
  ptr += __shfl_xor(ptr, 2, 32);
  ptr += __shfl_xor(ptr, 4, 32);
  ptr += __shfl_xor(ptr, 8, 32);
  ptr += __shfl_xor(ptr, 16, 32);
  if ((t & 31) == 0) str[t >> 5] = ptr;
  __syncthreads();
  if (t == 0) {
    float s = 0.0f;
    for (int i = 0; i < 8; ++i) s += str[i];
    atomicAdd(&scal[1], s);
  }
  if (t < K_DIM) atomicAdd(&nl[t], snl[t]);
}

// --------------- P[16,128] = S^T @ X  (WMMA, atomic accumulate) ------------
// Wave w of each block owns channel group w; block owns a node range.
__global__ void pool_kernel(const float* __restrict__ assign,
                            const float* __restrict__ feat,
                            float* __restrict__ P, int N) {
  int w = threadIdx.x >> 5;  // channel group 0..7
  int lane = threadIdx.x & 31;
  int col = lane & 15;   // A: cluster j ; B/D: channel within group
  int half = lane >> 4;
  int cbase = w * 16;
  int npb = N / gridDim.x;  // nodes per block (N divisible by grid)
  int n0 = blockIdx.x * npb;
  int n1 = n0 + npb;
  v8f acc = {0.f, 0.f, 0.f, 0.f, 0.f, 0.f, 0.f, 0.f};
  for (int nb = n0; nb < n1; nb += 4) {
    int i0 = nb + half * 2;
    v2f a, b;
    a.x = assign[(size_t)i0 * K_DIM + col];         // A[j][k0]   = S[i0][j]
    a.y = assign[(size_t)(i0 + 1) * K_DIM + col];   // A[j][k0+1] = S[i0+1][j]
    b.x = feat[(size_t)i0 * C_DIM + cbase + col];   // B[k0][c]
    b.y = feat[(size_t)(i0 + 1) * C_DIM + cbase + col];
    acc = wmma_f32_4(a, b, acc);
  }
#pragma unroll
  for (int v = 0; v < 8; ++v) {
    int j = v + half * 8;
    atomicAdd(&P[(size_t)j * C_DIM + cbase + col], acc[v]);
  }
}

// ---------------- Ppost = selu(P / cluster_sizes[:,None]) ------------------
__global__ void selu_kernel(const float* __restrict__ P,
                            const float* __restrict__ cs,
                            float* __restrict__ Pp) {
  int t = blockIdx.x * blockDim.x + threadIdx.x;
  if (t < K_DIM * C_DIM) {
    int j = t >> 7;
    float x = P[t] / cs[j];
    const float scale = 1.0507009873554805f;
    const float alpha = 1.6732632423543772f;
    Pp[t] = x > 0.0f ? scale * x : scale * alpha * (__expf(x) - 1.0f);
  }
}

// ------------------------------- scalar loss -------------------------------
__global__ void loss_kernel(const float* __restrict__ scal,
                            const float* __restrict__ nl,
                            const float* __restrict__ cs,
                            float* __restrict__ out_loss, int N) {
  if (threadIdx.x == 0 && blockIdx.x == 0) {
    float twoE = 2.0f * scal[0];
    float tr_gp = scal[1];
    float tn = 0.0f;
    for (int j = 0; j < K_DIM; ++j) tn += nl[j] * nl[j];
    tn /= twoE;
    float spectral = -(tr_gp - tn) / twoE;
    float target = (float)N / (float)K_DIM;
    float cl = 0.0f;
    for (int j = 0; j < K_DIM; ++j) cl += fabsf(cs[j] - target);
    float sk = sqrtf((float)K_DIM);
    cl = cl / (float)N * sk / (sk - 1.0f) * 0.5f;
    *out_loss = spectral + cl;
  }
}

// ----------- out[N,128] = (S / cs) @ Ppost   (WMMA unpool) -----------------
// One block per 16-node tile; wave w owns channel group w. K loop = 16.
__global__ void unpool_kernel(const float* __restrict__ assign,
                              const float* __restrict__ cs,
                              const float* __restrict__ Pp,
                              float* __restrict__ outf) {
  int tile = blockIdx.x;
  int w = threadIdx.x >> 5;
  int lane = threadIdx.x & 31;
  int col = lane & 15;   // A: node row m ; B/D: channel within group
  int half = lane >> 4;
  int cbase = w * 16;
  int ko = half * 2;
  const float* arow = assign + (size_t)(tile * 16 + col) * K_DIM;
  v8f acc = {0.f, 0.f, 0.f, 0.f, 0.f, 0.f, 0.f, 0.f};
#pragma unroll
  for (int kb = 0; kb < K_DIM; kb += 4) {
    int k0 = kb + ko;
    v2f a, b;
    a.x = arow[k0] / cs[k0];           // assignments_pooling[m][k0]
    a.y = arow[k0 + 1] / cs[k0 + 1];
    b.x = Pp[(size_t)k0 * C_DIM + cbase + col];
    b.y = Pp[(size_t)(k0 + 1) * C_DIM + cbase + col];
    acc = wmma_f32_4(a, b, acc);
  }
#pragma unroll
  for (int v = 0; v < 8; ++v) {
    int r = v + half * 8;
    outf[(size_t)(tile * 16 + r) * C_DIM + cbase + col] = acc[v];
  }
}

// ---------------------------------------------------------------------------
extern "C" void kernel_launch(void* const* d_in, const int* in_sizes, int n_in,
                              void* d_out, int out_size, void* d_ws,
                              size_t ws_size, hipStream_t stream) {
  const float* feat = (const float*)d_in[0];  // [N,128]
  const int* erow = (const int*)d_in[1];      // [E]
  const int* ecol = (const int*)d_in[2];      // [E]
  const float* eval = (const float*)d_in[3];  // [E]
  const float* fcw = (const float*)d_in[4];   // [16,128]
  const float* fcb = (const float*)d_in[5];   // [16]

  int N = in_sizes[0] / C_DIM;  // 100000
  int E = in_sizes[1];          // 3200000

  float* out = (float*)d_out;
  float* out_assign = out + (size_t)N * C_DIM;
  float* out_loss = out_assign + (size_t)N * K_DIM;

  // workspace layout
  float* ws = (float*)d_ws;
  float* a_s = ws;                               // N*16
  float* deg = a_s + (size_t)N * K_DIM;          // N
  float* cs = deg + N;                           // 16
  float* nl = cs + K_DIM;                        // 16
  float* scal = nl + K_DIM;                      // 8: [0]=numEdges [1]=tr(S^T A S)
  float* P = scal + 8;                           // 16*128
  float* Pp = P + K_DIM * C_DIM;                 // 16*128

  long long zn = (long long)N * K_DIM + N + K_DIM + K_DIM + 8 +
                 2LL * K_DIM * C_DIM;
  zero_kernel<<<1024, 256, 0, stream>>>(ws, zn);

  int nTiles = N / 16;  // 6250 (N divides exactly)
  assign_kernel<<<nTiles / 2, 64, 0, stream>>>(feat, fcw, fcb, out_assign,
                                               nTiles);
  cs_kernel<<<128, 256, 0, stream>>>(out_assign, cs, N);
  deg_kernel<<<512, 256, 0, stream>>>(ecol, eval, deg, scal, E);

  long long spthreads = (long long)E * 16;
  spmm_kernel<<<(int)((spthreads + 255) / 256), 256, 0, stream>>>(
      erow, ecol, eval, out_assign, a_s, E);

  trace_kernel<<<128, 256, 0, stream>>>(a_s, out_assign, deg, nl, scal, N);

  pool_kernel<<<250, 256, 0, stream>>>(out_assign, feat, P, N);  // 100000/250=400
  selu_kernel<<<(K_DIM * C_DIM + 255) / 256, 256, 0, stream>>>(P, cs, Pp);
  loss_kernel<<<1, 32, 0, stream>>>(scal, nl, cs, out_loss, N);
  unpool_kernel<<<nTiles, 256, 0, stream>>>(out_assign, cs, Pp, out);
}